// IVCML_19971597927218
// MI455X (gfx1250) — compile-verified
//
#include <hip/hip_runtime.h>

#define B_ 8
#define N_ 2048
#define D_ 128
#define MAX_STEPS 4
#define WPAD_ROWS 16   // A-matrix padded to 16 rows; rows 8..15 are zero

typedef __attribute__((ext_vector_type(2))) float v2f;
typedef __attribute__((ext_vector_type(8))) float v8f;

// ---------------------------------------------------------------------------
// Init: cover = (s>0), frontier0 = cover, w = cover * alpha (seed weight a^1).
// Grid spans WPAD_ROWS*N_ so the padded w rows 8..15 are zeroed here once,
// making the WMMA A-fragment loads unconditional (no exec-mask predication).
// ---------------------------------------------------------------------------
__global__ void ivcml_init_kernel(const float* __restrict__ s,
                                  const float* __restrict__ alpha_p,
                                  float* __restrict__ cover,
                                  float* __restrict__ frontier,
                                  float* __restrict__ w) {
    int idx = blockIdx.x * blockDim.x + threadIdx.x;   // < WPAD_ROWS*N_
    if (idx < B_ * N_) {
        float alpha = *alpha_p;
        float c = (s[idx] > 0.f) ? 1.f : 0.f;
        cover[idx]    = c;
        frontier[idx] = c;
        w[idx]        = c * alpha;
    } else {
        w[idx] = 0.f;                                  // zero pad rows 8..15
    }
}

// ---------------------------------------------------------------------------
// One BFS step. Grid: B_ * (N_/256) blocks of 256 threads.
// Frontier of batch b staged into LDS as a 2048-bit mask; each thread owns one
// destination column o, scans only frontier rows (coalesced across o), and
// stops at the first incoming edge. Newly covered nodes get weight a^(step+2).
// ---------------------------------------------------------------------------
__global__ void ivcml_step_kernel(const float* __restrict__ a,
                                  const float* __restrict__ alpha_p,
                                  const int*   __restrict__ step_num_p,
                                  const float* __restrict__ frontier_in,
                                  float*       __restrict__ frontier_out,
                                  float*       __restrict__ cover,
                                  float*       __restrict__ w,
                                  int step) {
    if (step >= *step_num_p) return;

    __shared__ unsigned mask[N_ / 32];   // 64 words = 2048 bits
    const int b     = blockIdx.x >> 3;   // N_/256 == 8 chunks per batch
    const int chunk = blockIdx.x & 7;
    const int tid   = threadIdx.x;

    if (tid < N_ / 32) {
        unsigned mword = 0u;
        const float* fr = frontier_in + (size_t)b * N_ + tid * 32;
        #pragma unroll 4
        for (int bit = 0; bit < 32; ++bit)
            if (fr[bit] > 0.f) mword |= (1u << bit);
        mask[tid] = mword;
    }
    __syncthreads();

    const int    o  = chunk * 256 + tid;
    const size_t co = (size_t)b * N_ + o;

    float fout = 0.f;
    if (cover[co] <= 0.f) {
        const float* ab = a + (size_t)b * N_ * N_;
        bool found = false;
        for (int j = 0; j < N_ / 32 && !found; ++j) {
            unsigned mword = mask[j];
            while (mword) {
                int bit = __ffs(mword) - 1;
                mword &= (mword - 1u);
                int i = j * 32 + bit;
                if (ab[(size_t)i * N_ + o] > 0.f) { found = true; break; }
            }
        }
        if (found) {
            float alpha = *alpha_p;
            float wv = alpha * alpha;            // a^2 for step 0
            for (int t = 0; t < step; ++t) wv *= alpha;
            cover[co] = 1.f;
            w[co]     = wv;
            fout      = 1.f;
        }
    }
    frontier_out[co] = fout;
}

// ---------------------------------------------------------------------------
// denom[b] = sum(cover[b,:])  (or 1.0 if step_num == 0, matching reference)
// ---------------------------------------------------------------------------
__global__ void ivcml_denom_kernel(const float* __restrict__ cover,
                                   const int*   __restrict__ step_num_p,
                                   float*       __restrict__ denom) {
    __shared__ float red[256];
    const int b   = blockIdx.x;
    const int tid = threadIdx.x;
    float sum = 0.f;
    for (int i = tid; i < N_; i += 256) sum += cover[(size_t)b * N_ + i];
    red[tid] = sum;
    __syncthreads();
    for (int st = 128; st > 0; st >>= 1) {
        if (tid < st) red[tid] += red[tid + st];
        __syncthreads();
    }
    if (tid == 0) denom[b] = (*step_num_p > 0) ? red[0] : 1.f;
}

// ---------------------------------------------------------------------------
// out[b,d] = (sum_k w[b,k] * fea_emb[k,d]) / denom[b]  via V_WMMA_F32_16X16X4_F32
// One 256-thread block = 8 waves; wave wv owns output columns [wv*16, wv*16+16).
// A = padded w (16 rows, rows 8..15 pre-zeroed) -> unconditional b64 loads.
// A-fragment: lane l (/ l+16) holds A[M=l, K=2h], A[M=l, K=2h+1]; B mirrors
// with N across lanes; D VGPR r = rows r (lanes 0-15) / r+8 (lanes 16-31).
// ---------------------------------------------------------------------------
__global__ void __launch_bounds__(256)
ivcml_gemm_kernel(const float* __restrict__ fea_emb,
                  const float* __restrict__ w,
                  const float* __restrict__ denom,
                  float* __restrict__ out) {
    const int tid   = threadIdx.x;
    const int wv_id = tid >> 5;          // 0..7 -> 16-wide column tile
    const int lane  = tid & 31;
    const int half  = lane >> 4;         // 0: K 0..1 / rows 0..7 ; 1: K 2..3 / rows 8..15
    const int l     = lane & 15;
    const int n     = wv_id * 16 + l;    // output column (0..127)

    const float* wrow = w + (size_t)l * N_ + 2 * half;   // this lane's A row
    const float* fcol = fea_emb + (size_t)(2 * half) * D_ + n;

    v8f c = {};
    #pragma unroll 8
    for (int k = 0; k < N_; k += 4) {
        v2f av = *(const v2f*)(wrow + k);        // 8B-aligned: k+2*half is even
        v2f bv;
        bv.x = fcol[(size_t)k * D_];
        bv.y = fcol[(size_t)k * D_ + D_];
        c = __builtin_amdgcn_wmma_f32_16x16x4_f32(
                /*neg_a=*/false, av, /*neg_b=*/false, bv,
                /*c_mod=*/(short)0, c, /*reuse_a=*/false, /*reuse_b=*/false);
    }

    if (half == 0) {
        #pragma unroll
        for (int r = 0; r < B_; ++r)     // D VGPR r holds row M=r in lanes 0..15
            out[(size_t)r * D_ + n] = c[r] / denom[r];
    }
}

// ---------------------------------------------------------------------------
extern "C" void kernel_launch(void* const* d_in, const int* in_sizes, int n_in,
                              void* d_out, int out_size, void* d_ws, size_t ws_size,
                              hipStream_t stream) {
    const float* a          = (const float*)d_in[0];   // [B,N,N]
    const float* s          = (const float*)d_in[1];   // [B,N]
    const float* fea_emb    = (const float*)d_in[2];   // [N,D]
    const float* alpha_p    = (const float*)d_in[3];   // scalar
    const int*   step_num_p = (const int*)  d_in[4];   // scalar
    float*       out        = (float*)d_out;           // [B,D]

    float* cover = (float*)d_ws;                    // B*N
    float* f0    = cover + (size_t)B_ * N_;         // B*N
    float* f1    = f0    + (size_t)B_ * N_;         // B*N
    float* w     = f1    + (size_t)B_ * N_;         // WPAD_ROWS*N (rows 8..15 zero)
    float* denom = w     + (size_t)WPAD_ROWS * N_;  // B

    ivcml_init_kernel<<<(WPAD_ROWS * N_) / 256, 256, 0, stream>>>(s, alpha_p, cover, f0, w);

    float* fin = f0;
    float* fout = f1;
    for (int stp = 0; stp < MAX_STEPS; ++stp) {
        ivcml_step_kernel<<<B_ * (N_ / 256), 256, 0, stream>>>(
            a, alpha_p, step_num_p, fin, fout, cover, w, stp);
        float* t = fin; fin = fout; fout = t;
    }

    ivcml_denom_kernel<<<B_, 256, 0, stream>>>(cover, step_num_p, denom);
    ivcml_gemm_kernel<<<1, 256, 0, stream>>>(fea_emb, w, denom, out);
}